// semantic_estimator_47802986004511
// MI455X (gfx1250) — compile-verified
//
#include <hip/hip_runtime.h>

typedef __attribute__((ext_vector_type(2))) float v2f;
typedef __attribute__((ext_vector_type(8))) float v8f;

#define A_DIM 256     // features per row
#define C_DIM 19      // classes
#define CPAD  32      // padded class count (two 16-row WMMA tiles)

// ---------------- workspace zeroing ----------------
__global__ __launch_bounds__(256) void zero_ws_kernel(float* __restrict__ ws, int n) {
  int i = blockIdx.x * blockDim.x + threadIdx.x;
  int stride = gridDim.x * blockDim.x;
  for (; i < n; i += stride) ws[i] = 0.0f;
}

// ---------------- label histogram (counts) ----------------
__global__ __launch_bounds__(256) void label_hist_kernel(const int* __restrict__ labels, int n,
                                                         float* __restrict__ cnt) {
  __shared__ int h[CPAD];
  if (threadIdx.x < CPAD) h[threadIdx.x] = 0;
  __syncthreads();
  for (int i = blockIdx.x * blockDim.x + threadIdx.x; i < n; i += gridDim.x * blockDim.x) {
    int l = labels[i];
    if ((unsigned)l < (unsigned)CPAD) atomicAdd(&h[l], 1);   // label 255 (IGNORE) self-masks
  }
  __syncthreads();
  if (threadIdx.x < CPAD) {
    int v = h[threadIdx.x];
    if (v != 0) unsafeAtomicAdd(&cnt[threadIdx.x], (float)v);
  }
}

// One 4-row chunk: 4x V_WMMA_F32_16X16X4_F32 (sum & sumsq x 2 class tiles).
__device__ __forceinline__ void chunk_wmma(float fa, float fb, int la, int lb, int m,
                                           v8f& s0, v8f& s1, v8f& q0, v8f& q1) {
  v2f b;  b.x  = fa;      b.y  = fb;
  v2f b2; b2.x = fa * fa; b2.y = fb * fb;
  v2f a0; a0.x = (la == m)      ? 1.0f : 0.0f; a0.y = (lb == m)      ? 1.0f : 0.0f;
  v2f a1; a1.x = (la == m + 16) ? 1.0f : 0.0f; a1.y = (lb == m + 16) ? 1.0f : 0.0f;
  s0 = __builtin_amdgcn_wmma_f32_16x16x4_f32(false, a0, false, b,  (short)0, s0, false, false);
  s1 = __builtin_amdgcn_wmma_f32_16x16x4_f32(false, a1, false, b,  (short)0, s1, false, false);
  q0 = __builtin_amdgcn_wmma_f32_16x16x4_f32(false, a0, false, b2, (short)0, q0, false, false);
  q1 = __builtin_amdgcn_wmma_f32_16x16x4_f32(false, a1, false, b2, (short)0, q1, false, false);
}

// ---------------- WMMA moment reduction: sum & sum-of-squares per class ----------------
// Per wave: one 16-column tile. A = one-hot(labels) for a 16-class tile (lane m = class,
// K = 2*(lane/16)+j), B = 4x16 feature chunk (lane m = column, same K mapping).
__global__ __launch_bounds__(512) void moment_reduce_kernel(const float* __restrict__ feat,
                                                            const int* __restrict__ labels,
                                                            int n, int rowsPerBlock,
                                                            float* __restrict__ accSum,
                                                            float* __restrict__ accSq) {
  const int lane = threadIdx.x & 31;
  const int wave = threadIdx.x >> 5;        // 0..15 -> column tile
  const int col0 = wave << 4;               // 16 columns per wave
  const int m    = lane & 15;               // A: class-in-tile | B/D: column-in-tile
  const int ktop = (lane >> 4) << 1;        // K base for this lane half (0 or 2)

  const int rowStart = blockIdx.x * rowsPerBlock;
  int rowEnd = rowStart + rowsPerBlock;
  if (rowEnd > n) rowEnd = n;
  const int rows   = rowEnd - rowStart;
  const int chunks = rows >> 2;             // full, unguarded 4-row chunks

  v8f s0 = {}; v8f s1 = {};                 // class tiles 0-15 / 16-31, sum
  v8f q0 = {}; v8f q1 = {};                 // class tiles 0-15 / 16-31, sum of squares

  // Steady state: pointer-increment addressing, immediate offsets, 2 chunks/iter.
  const float* pa = feat + (size_t)(rowStart + ktop) * A_DIM + col0 + m;
  const int*   pl = labels + rowStart + ktop;

  int i = 0;
  for (; i + 2 <= chunks; i += 2) {
    float fa0 = pa[0];
    float fb0 = pa[A_DIM];
    float fa1 = pa[4 * A_DIM];
    float fb1 = pa[5 * A_DIM];
    int   la0 = pl[0], lb0 = pl[1];
    int   la1 = pl[4], lb1 = pl[5];

    // block-uniform guard -> scalar branch, no EXEC churn; ~16 chunks (64 rows) ahead
    if (i + 18 <= chunks)
      __builtin_prefetch(pa + 64 * A_DIM, 0, 1);

    chunk_wmma(fa0, fb0, la0, lb0, m, s0, s1, q0, q1);
    chunk_wmma(fa1, fb1, la1, lb1, m, s0, s1, q0, q1);

    pa += 8 * A_DIM;
    pl += 8;
  }
  if (i < chunks) {                         // odd leftover full chunk
    chunk_wmma(pa[0], pa[A_DIM], pl[0], pl[1], m, s0, s1, q0, q1);
    pa += 4 * A_DIM;
    pl += 4;
  }

  // Guarded tail (rows % 4 != 0): clamp addresses, zero one-hot for invalid rows.
  const int base = rowStart + (chunks << 2);
  if (base < rowEnd) {
    int ra  = base + ktop;
    int rb  = ra + 1;
    int rac = (ra < n) ? ra : (n - 1);
    int rbc = (rb < n) ? rb : (n - 1);
    int la  = (ra < rowEnd) ? labels[rac] : -1;
    int lb  = (rb < rowEnd) ? labels[rbc] : -1;
    float fa = feat[(size_t)rac * A_DIM + col0 + m];
    float fb = feat[(size_t)rbc * A_DIM + col0 + m];
    chunk_wmma(fa, fb, la, lb, m, s0, s1, q0, q1);
  }

  // Flush per-wave partials. D layout: VGPR v, lane l -> M = v + 8*(l/16), N = l%16.
  const int nd   = lane & 15;
  const int mtop = (lane >> 4) << 3;
#pragma unroll
  for (int v = 0; v < 8; ++v) {
    int mc = mtop + v;
    unsafeAtomicAdd(&accSum[(size_t)mc * A_DIM + col0 + nd],        s0[v]);
    unsafeAtomicAdd(&accSum[(size_t)(mc + 16) * A_DIM + col0 + nd], s1[v]);
    unsafeAtomicAdd(&accSq [(size_t)mc * A_DIM + col0 + nd],        q0[v]);
    unsafeAtomicAdd(&accSq [(size_t)(mc + 16) * A_DIM + col0 + nd], q1[v]);
  }
}

// ---------------- finalize: EMA merge + pack [CoV | Mean | Amount] ----------------
__global__ __launch_bounds__(256) void finalize_kernel(const float* __restrict__ Mean,
                                                       const float* __restrict__ CoV,
                                                       const float* __restrict__ Amount,
                                                       const float* __restrict__ accSum,
                                                       const float* __restrict__ accSq,
                                                       const float* __restrict__ cnt,
                                                       float* __restrict__ out) {
  const int c = blockIdx.x;      // 0..18
  const int a = threadIdx.x;     // 0..255
  float count = cnt[c];
  float cc    = fmaxf(count, 1.0f);
  float s     = accSum[(size_t)c * A_DIM + a];
  float q     = accSq [(size_t)c * A_DIM + a];
  float mean  = s / cc;
  float var   = fmaxf(q / cc - mean * mean, 0.0f);   // == two-pass var when count>=1

  float am    = Amount[c];
  float denom = count + am;
  float w     = (denom > 0.0f) ? (count / denom) : 0.0f;   // NaN (0/0) -> 0

  float M0 = Mean[(size_t)c * A_DIM + a];
  float V0 = CoV [(size_t)c * A_DIM + a];
  float dm = M0 - mean;
  float addCV = w * (1.0f - w) * dm * dm;

  const size_t base = (size_t)c * (2 * A_DIM + 1);
  out[base + a]         = V0 * (1.0f - w) + var * w + addCV;  // new_CoV
  out[base + A_DIM + a] = M0 * (1.0f - w) + mean * w;         // new_Mean
  if (a == 0) out[base + 2 * A_DIM] = am + count;             // new_Amount
}

// ---------------- launch ----------------
extern "C" void kernel_launch(void* const* d_in, const int* in_sizes, int n_in,
                              void* d_out, int out_size, void* d_ws, size_t ws_size,
                              hipStream_t stream) {
  const float* feat   = (const float*)d_in[0];
  const int*   labels = (const int*)  d_in[1];
  const float* Mean   = (const float*)d_in[2];
  const float* CoV    = (const float*)d_in[3];
  const float* Amount = (const float*)d_in[4];
  float*       out    = (float*)d_out;

  const int N = in_sizes[1];   // number of rows (labels count)

  float* ws     = (float*)d_ws;
  float* accSum = ws;                           // [32][256]
  float* accSq  = ws + CPAD * A_DIM;            // [32][256]
  float* cnt    = ws + 2 * CPAD * A_DIM;        // [32]
  const int wsFloats = 2 * CPAD * A_DIM + CPAD;

  zero_ws_kernel<<<32, 256, 0, stream>>>(ws, wsFloats);
  label_hist_kernel<<<64, 256, 0, stream>>>(labels, N, cnt);

  const int G = 256;                                         // reduction blocks
  const int rowsPerBlock = ((N + G * 4 - 1) / (G * 4)) * 4;  // multiple of 4
  moment_reduce_kernel<<<G, 512, 0, stream>>>(feat, labels, N, rowsPerBlock, accSum, accSq);

  finalize_kernel<<<C_DIM, A_DIM, 0, stream>>>(Mean, CoV, Amount, accSum, accSq, cnt, out);
}